// EqPBC_43508018709154
// MI455X (gfx1250) — compile-verified
//
#include <hip/hip_runtime.h>
#include <math.h>

// ---------------------------------------------------------------------------
// Problem constants (match reference)
// ---------------------------------------------------------------------------
#define MLEN   41
#define LHALF  20
#define THRESH 20
#define NMODES 2
#define HDIM   177

#define WAVES_PER_BLOCK  4
#define THREADS_PER_BLOCK (WAVES_PER_BLOCK * 32)
#define BATCH_PER_WAVE   16
#define BATCH_PER_BLOCK  (WAVES_PER_BLOCK * BATCH_PER_WAVE)
// Padded per-batch LDS stride in floats: 43*4 = 172.
//  - multiple of 4 -> every float4 row load is 16B aligned (ds_load_b128)
//  - 172 mod 64 = 44, gcd(44,64)=4 -> 16 consecutive lanes hit 16 distinct banks
#define SB 172

typedef __attribute__((ext_vector_type(2))) float v2f;
typedef __attribute__((ext_vector_type(8))) float v8f;

// ---------------------------------------------------------------------------
// Compile-time (m,n) index table, one packed u32 per h:
//   bits [7:0]   = L+n        (E index for En)
//   bits [15:8]  = L+m        (E index for Em)
//   bits [23:16] = L+m+n      (E index for conj term; JAX wrap/clamp applied)
//   bits [31:24] = (m == n)
// JAX gather semantics: negative indices wrap (only (-20,-1): -1 -> 40),
// out-of-range clamps (only (1,20): 41 -> 40).
// ---------------------------------------------------------------------------
struct IdxTab { unsigned int p[HDIM]; };

constexpr int count_pairs() {
    int h = 0;
    for (int m = -LHALF; m <= LHALF; ++m)
        for (int n = m; n <= LHALF; ++n) {
            int prod = m * n; if (prod < 0) prod = -prod;
            if (prod <= THRESH) ++h;
        }
    return h;
}
static_assert(count_pairs() == HDIM, "HDIM mismatch");

constexpr IdxTab make_tab() {
    IdxTab t{};
    int h = 0;
    for (int m = -LHALF; m <= LHALF; ++m)
        for (int n = m; n <= LHALF; ++n) {
            int prod = m * n; if (prod < 0) prod = -prod;
            if (prod > THRESH) continue;
            int e = LHALF + m + n;
            if (e < 0) e += MLEN;            // numpy/JAX negative wrap
            if (e > MLEN - 1) e = MLEN - 1;  // JAX OOB clamp
            unsigned int en  = (unsigned int)(LHALF + n);
            unsigned int em  = (unsigned int)(LHALF + m);
            unsigned int emn = (unsigned int)e;
            unsigned int dg  = (unsigned int)(m == n);
            t.p[h] = en | (em << 8) | (emn << 16) | (dg << 24);
            ++h;
        }
    return t;
}
__constant__ IdxTab c_tab = make_tab();

// ---------------------------------------------------------------------------
// Kernel: one wave handles 16 batches; the h-contraction runs as 177 chained
// V_WMMA_F32_16X16X4_F32 steps (K=4 per h: mode0{re,im}, mode1{re,im}).
// ---------------------------------------------------------------------------
__global__ __launch_bounds__(THREADS_PER_BLOCK)
void eqpbc_wmma_kernel(const float* __restrict__ xr,
                       const float* __restrict__ xi,
                       const float* __restrict__ task_info,
                       const float* __restrict__ Wr,
                       const float* __restrict__ Wi,
                       float* __restrict__ out,
                       int nbatch)
{
    // E tile: [b_local][pos][mode][re/im], padded stride SB floats per batch
    __shared__ __align__(16) float sE[BATCH_PER_BLOCK * SB];
    __shared__ __align__(8) float sWp[NMODES * HDIM * 2]; // {Wr,Wi} pairs, [mode][h]
    __shared__ float sD[WAVES_PER_BLOCK][16][17];    // accumulator spill, padded

    const int tid = threadIdx.x;
    const int b0  = blockIdx.x * BATCH_PER_BLOCK;
    const int gmax = nbatch * (MLEN * NMODES) - 1;

    // ---- stage W as interleaved pairs: one ds_load_b64 per iteration later ----
    for (int i = tid; i < NMODES * HDIM; i += THREADS_PER_BLOCK) {
        sWp[2 * i]     = Wr[i];
        sWp[2 * i + 1] = Wi[i];
    }
    // ---- stage E, interleaving re/im: coalesced global reads ----
    for (int i = tid; i < BATCH_PER_BLOCK * MLEN * NMODES; i += THREADS_PER_BLOCK) {
        int bl  = i / (MLEN * NMODES);
        int rem = i - bl * (MLEN * NMODES);          // pos*2 + mode
        int g   = (b0 + bl) * (MLEN * NMODES) + rem;
        g = g > gmax ? gmax : g;                     // tail clamp (loads only)
        float re = xr[g], im = xi[g];
        int o = bl * SB + rem * 2;
        sE[o]     = re;
        sE[o + 1] = im;
    }
    __syncthreads();

    const int wave = tid >> 5;
    const int lane = tid & 31;
    const int row  = lane & 15;    // A row (batch in tile) == B column index N
    const int mode = lane >> 4;    // low half-wave: K0/K1 (mode0); high: K2/K3 (mode1)
    const float* base = &sE[(wave * BATCH_PER_WAVE + row) * SB];
    const float2* wq  = (const float2*)&sWp[2 * mode * HDIM];

    // Loop-invariant B-column masks: this half-wave's output columns are
    // 2*mode (real) and 2*mode+1 (imag).
    const float m0 = (row == 2 * mode)     ? 1.0f : 0.0f;
    const float m1 = (row == 2 * mode + 1) ? 1.0f : 0.0f;

    v8f acc = {0.f, 0.f, 0.f, 0.f, 0.f, 0.f, 0.f, 0.f};

    for (int h = 0; h < HDIM; ++h) {
        const unsigned int pk = c_tab.p[h];          // 1x s_load_b32 + SALU unpack
        const int ien  = (int)(pk & 0xffu);
        const int iem  = (int)((pk >> 8) & 0xffu);
        const int iemn = (int)((pk >> 16) & 0xffu);

        // [re0, im0, re1, im1] quads, 16B-aligned ds_load_b128
        const float4 En = *(const float4*)(base + 4 * ien);
        const float4 Em = *(const float4*)(base + 4 * iem);
        const float4 Ec = *(const float4*)(base + 4 * iemn);
        // this half-wave's {Wr,Wi} pair: one ds_load_b64
        const float2 w  = wq[h];

        // s1 = sum_modes En * conj(Ec);  s2 = sum_modes Em * conj(Ec)
        float s1r = En.x * Ec.x + En.y * Ec.y + En.z * Ec.z + En.w * Ec.w;
        float s1i = En.y * Ec.x - En.x * Ec.y + En.w * Ec.z - En.z * Ec.w;
        float s2r = Em.x * Ec.x + Em.y * Ec.y + Em.z * Ec.z + Em.w * Ec.w;
        float s2i = Em.y * Ec.x - Em.x * Ec.y + Em.w * Ec.z - Em.z * Ec.w;

        // own-mode Em / En
        float emr = mode ? Em.z : Em.x, emi = mode ? Em.w : Em.y;
        float enr = mode ? En.z : En.x, eni = mode ? En.w : En.y;

        float tr = s1r * emr - s1i * emi;
        float ti = s1r * emi + s1i * emr;
        if (!(pk >> 24)) {                  // uniform (SALU) branch on diag
            tr += s2r * enr - s2i * eni;
            ti += s2r * eni + s2i * enr;
        }

        // A fragment: K even -> feat_re, K odd -> feat_im (this lane's mode)
        v2f a; a.x = tr; a.y = ti;

        // B fragment: complex-multiply structure against W[mode][h]
        //   col 2m   (out_re): rows {re,im} carry { Wr, -Wi }
        //   col 2m+1 (out_im): rows {re,im} carry { Wi,  Wr }
        v2f b;
        b.x = m0 * w.x + m1 * w.y;
        b.y = m1 * w.x - m0 * w.y;

        acc = __builtin_amdgcn_wmma_f32_16x16x4_f32(
            false, a, false, b, (short)0, acc, false, false);
    }

    // Spill D (16x16 f32 layout: VGPR v -> M = v + 8*(lane>>4), N = lane&15)
    {
        const int mb = mode * 8;
        #pragma unroll
        for (int v = 0; v < 8; ++v)
            sD[wave][mb + v][row] = acc[v];
    }
    __syncthreads();

    // Epilogue: one thread per batch combines center tap + P * Eout
    if (tid < BATCH_PER_BLOCK) {
        const int b = b0 + tid;
        if (b < nbatch) {
            const int w = tid >> 4, r = tid & 15;
            float d0 = sD[w][r][0], d1 = sD[w][r][1];
            float d2 = sD[w][r][2], d3 = sD[w][r][3];
            // P = 10^(task/10) / NMODES
            float P = 0.5f * __expf(task_info[b * 4] * 0.23025850929940457f);
            const float* eb = &sE[tid * SB + 4 * LHALF];   // E[:, L, :]
            float4 o;
            o.x = eb[0] + P * d0;   // mode0 re
            o.y = eb[1] + P * d1;   // mode0 im
            o.z = eb[2] + P * d2;   // mode1 re
            o.w = eb[3] + P * d3;   // mode1 im
            *(float4*)(out + (size_t)b * 4) = o;
        }
    }
}

// ---------------------------------------------------------------------------
// Launch
// ---------------------------------------------------------------------------
extern "C" void kernel_launch(void* const* d_in, const int* in_sizes, int n_in,
                              void* d_out, int out_size, void* d_ws, size_t ws_size,
                              hipStream_t stream) {
    const float* xr = (const float*)d_in[0];
    const float* xi = (const float*)d_in[1];
    const float* ti = (const float*)d_in[2];
    const float* Wr = (const float*)d_in[3];
    const float* Wi = (const float*)d_in[4];
    float* out = (float*)d_out;

    const int nbatch  = in_sizes[0] / (MLEN * NMODES);     // 65536
    const int nblocks = (nbatch + BATCH_PER_BLOCK - 1) / BATCH_PER_BLOCK;

    eqpbc_wmma_kernel<<<nblocks, THREADS_PER_BLOCK, 0, stream>>>(
        xr, xi, ti, Wr, Wi, out, nbatch);
}